// GCNEncoder_78494822302121
// MI455X (gfx1250) — compile-verified
//
#include <hip/hip_runtime.h>

#define N_NODES 100000
#define N_EDGES 3200000
#define IN_DIM 128
#define HID_DIM 64
#define EMB_DIM 32

typedef __attribute__((ext_vector_type(2))) float v2f;
typedef __attribute__((ext_vector_type(8))) float v8f;

// ---------------- degree / normalization ----------------

__global__ void deg_init_kernel(float* __restrict__ deg, int n) {
    int i = blockIdx.x * blockDim.x + threadIdx.x;
    if (i < n) deg[i] = 1.0f;  // self-loop guarantees deg >= 1
}

__global__ void deg_accum_kernel(const int* __restrict__ ei, float* __restrict__ deg, int e_total) {
    int e = blockIdx.x * blockDim.x + threadIdx.x;
    if (e < e_total) atomicAdd(&deg[ei[N_EDGES + e]], 1.0f);  // dst row of edge_index
}

__global__ void deg_rsqrt_kernel(float* __restrict__ deg, int n) {
    int i = blockIdx.x * blockDim.x + threadIdx.x;
    if (i < n) deg[i] = rsqrtf(deg[i]);
}

// ---------------- dense transform: Y[N,NOUT] = (relu?)X[N,K] @ X W[K,NOUT] ----------------
// One wave32 computes a 16-row x NOUT tile with V_WMMA_F32_16X16X4_F32.
// A layout (16x4 f32): lanes 0-15 hold K={k0,k0+1}, lanes 16-31 hold K={k0+2,k0+3}, M = lane&15.
// B layout (4x16 f32): VGPR0 = rows {k0, k0+2}, VGPR1 = rows {k0+1, k0+3}, N = lane&15.
// C/D (16x16 f32): VGPR j -> row j (lanes 0-15) / row j+8 (lanes 16-31), N = lane&15.
//
// W is staged in LDS *K-pair interleaved*: element (k,n) at [(k>>1)*2*NOUT + n*2 + (k&1)],
// so a B fragment {W[kk][n], W[kk+1][n]} is one aligned ds_load_b64 (no register shuffles).
template <int K, int NOUT, bool RELU_A>
__global__ __launch_bounds__(256) void gemm_wmma_kernel(const float* __restrict__ X,
                                                        const float* __restrict__ W,
                                                        float* __restrict__ Y, int n_rows) {
    constexpr int NT = NOUT / 16;
    __shared__ float Wlds[K * NOUT];
    const int tid = threadIdx.x;
    for (int i = tid; i < K * NOUT; i += 256) {
        const int k = i / NOUT, n = i % NOUT;
        Wlds[(k >> 1) * (2 * NOUT) + n * 2 + (k & 1)] = W[i];
    }
    __syncthreads();

    const int wave  = tid >> 5;
    const int lane  = tid & 31;
    const int m     = lane & 15;
    const int khalf = lane >> 4;  // 0 or 1
    const long row0 = (long)blockIdx.x * 128 + (long)wave * 16;

    long rA = row0 + m;
    if (rA > (long)(n_rows - 1)) rA = n_rows - 1;  // clamp so EXEC stays all-1s for WMMA
    const float* __restrict__ xrow = X + rA * K + 2 * khalf;
    // per-lane LDS base: column (t*16+m), K-pair row kp = (k0+2*khalf)>>1 = k0/2 + khalf
    const float* __restrict__ wbase = &Wlds[khalf * (2 * NOUT) + m * 2];

    v8f acc[NT] = {};

    for (int k0 = 0; k0 < K; k0 += 4) {
        v2f a;
        a.x = xrow[k0];
        a.y = xrow[k0 + 1];
        if (RELU_A) {
            a.x = fmaxf(a.x, 0.0f);
            a.y = fmaxf(a.y, 0.0f);
        }
        const float* __restrict__ wk = wbase + (k0 >> 1) * (2 * NOUT);
#pragma unroll
        for (int t = 0; t < NT; ++t) {
            v2f b = *(const v2f*)(wk + t * 32);  // {W[kk][n], W[kk+1][n]} in one b64 load
            acc[t] = __builtin_amdgcn_wmma_f32_16x16x4_f32(
                false, a, false, b, (short)0, acc[t], false, false);
        }
    }

#pragma unroll
    for (int j = 0; j < 8; ++j) {
        const long r = row0 + j + khalf * 8;
        if (r < n_rows) {
#pragma unroll
            for (int t = 0; t < NT; ++t)
                Y[r * NOUT + t * 16 + m] = acc[t][j];
        }
    }
}

// ---------------- aggregation init: out = bias + self-loop term ----------------
template <int D>
__global__ void agg_init_kernel(const float* __restrict__ h, const float* __restrict__ dinv,
                                const float* __restrict__ bias, float* __restrict__ out, int n) {
    long idx = (long)blockIdx.x * blockDim.x + threadIdx.x;
    if (idx < (long)n * D) {
        int i = (int)(idx / D);
        int c = (int)(idx % D);
        float di = dinv[i];
        out[idx] = bias[c] + h[idx] * di * di;
    }
}

// ---------------- edge scatter: one wave32 per edge, lanes cover feature dim ----------------
template <int D>
__global__ void edge_agg_kernel(const int* __restrict__ ei, const float* __restrict__ dinv,
                                const float* __restrict__ h, float* __restrict__ out) {
    const int lane = threadIdx.x & 31;
    const int wpb  = blockDim.x >> 5;
    long wave   = (long)blockIdx.x * wpb + (threadIdx.x >> 5);
    long nWaves = (long)gridDim.x * wpb;
    for (long e = wave; e < N_EDGES; e += nWaves) {
        const int s = ei[e];
        const int d = ei[N_EDGES + e];
        const float nrm = dinv[s] * dinv[d];
        const float* __restrict__ hs = h + (long)s * D;
        float* __restrict__ od = out + (long)d * D;
#pragma unroll
        for (int c = lane; c < D; c += 32)
            atomicAdd(&od[c], hs[c] * nrm);  // L2-resident f32 atomics (buffers fit in 192MB L2)
    }
}

// ---------------- host ----------------
extern "C" void kernel_launch(void* const* d_in, const int* in_sizes, int n_in,
                              void* d_out, int out_size, void* d_ws, size_t ws_size,
                              hipStream_t stream) {
    const float* x  = (const float*)d_in[0];
    const float* W1 = (const float*)d_in[1];
    const float* b1 = (const float*)d_in[2];
    const float* W2 = (const float*)d_in[3];
    const float* b2 = (const float*)d_in[4];
    const int*   ei = (const int*)d_in[5];

    float* ws     = (float*)d_ws;
    float* dinv   = ws;                                   // N floats
    float* hbuf   = ws + N_NODES;                         // N*64 (h1, then reused for t2)
    float* aggbuf = hbuf + (size_t)N_NODES * HID_DIM;     // N*64

    float* out = (float*)d_out;

    const int T = 256;
    // degree -> deg^-1/2
    deg_init_kernel<<<(N_NODES + T - 1) / T, T, 0, stream>>>(dinv, N_NODES);
    deg_accum_kernel<<<(N_EDGES + T - 1) / T, T, 0, stream>>>(ei, dinv, N_EDGES);
    deg_rsqrt_kernel<<<(N_NODES + T - 1) / T, T, 0, stream>>>(dinv, N_NODES);

    const int gemmGrid = (N_NODES + 127) / 128;

    // layer 1: h1 = x @ W1 ; agg = bias+selfloop ; scatter
    gemm_wmma_kernel<IN_DIM, HID_DIM, false><<<gemmGrid, 256, 0, stream>>>(x, W1, hbuf, N_NODES);
    agg_init_kernel<HID_DIM>
        <<<(int)(((long)N_NODES * HID_DIM + T - 1) / T), T, 0, stream>>>(hbuf, dinv, b1, aggbuf, N_NODES);
    edge_agg_kernel<HID_DIM><<<4096, 256, 0, stream>>>(ei, dinv, hbuf, aggbuf);

    // layer 2: t2 = relu(agg) @ W2 (relu fused into A loads) ; init out ; scatter
    gemm_wmma_kernel<HID_DIM, EMB_DIM, true><<<gemmGrid, 256, 0, stream>>>(aggbuf, W2, hbuf, N_NODES);
    agg_init_kernel<EMB_DIM>
        <<<(int)(((long)N_NODES * EMB_DIM + T - 1) / T), T, 0, stream>>>(hbuf, dinv, b2, out, N_NODES);
    edge_agg_kernel<EMB_DIM><<<4096, 256, 0, stream>>>(ei, dinv, hbuf, out);
}